// SCSA_15633680957553
// MI455X (gfx1250) — compile-verified
//
#include <hip/hip_runtime.h>
#include <math.h>

// ---- problem constants -------------------------------------------------
#define B_    32
#define C_    256
#define HH    56
#define LL    56
#define PL    3136   // 56*56
#define PL4   784    // PL/4
#define ROW4  14     // 56/4 float4 per row
#define PH    8      // pooled h
#define PN    64     // pooled n = 8*8
#define HEADS_ 8
#define HD    32     // head dim
#define EPSN  1e-5f

typedef __attribute__((ext_vector_type(2))) float v2f;
typedef __attribute__((ext_vector_type(8))) float v8f;

// ---- WMMA f32 16x16x4 step (D = A(16x4) * B(4x16) + C) -----------------
__device__ inline v8f wmma4(v2f a, v2f b, v8f c) {
#if __has_builtin(__builtin_amdgcn_wmma_f32_16x16x4_f32)
  return __builtin_amdgcn_wmma_f32_16x16x4_f32(false, a, false, b, (short)0, c,
                                               false, false);
#else
  return c;
#endif
}

__device__ inline float sig_(float v) { return 1.0f / (1.0f + expf(-v)); }

// ---- kernel 1: per-(b,c) row means (over w) and col means (over h) -----
// float4 streaming; a float4 never crosses a row (56 % 4 == 0), so the
// 4 row contributions fold into one LDS atomic.
__global__ void k_means(const float* __restrict__ x,
                        float* __restrict__ xh, float* __restrict__ xw) {
  int bc = blockIdx.x;                       // b*C + c
  const float4* p4 = reinterpret_cast<const float4*>(x + (size_t)bc * PL);
  __shared__ float rows[HH];
  __shared__ float cols[LL];
  int t = threadIdx.x;
  if (t < HH) { rows[t] = 0.f; cols[t] = 0.f; }
  __syncthreads();
  for (int idx4 = t; idx4 < PL4; idx4 += 256) {
    float4 v = p4[idx4];
    int i = idx4 / ROW4;
    int j = (idx4 - i * ROW4) * 4;
    atomicAdd(&rows[i], (v.x + v.y) + (v.z + v.w));
    atomicAdd(&cols[j + 0], v.x);
    atomicAdd(&cols[j + 1], v.y);
    atomicAdd(&cols[j + 2], v.z);
    atomicAdd(&cols[j + 3], v.w);
  }
  __syncthreads();
  if (t < HH) {
    xh[(size_t)bc * LL + t] = rows[t] * (1.f / 56.f);
    xw[(size_t)bc * LL + t] = cols[t] * (1.f / 56.f);
  }
}

// ---- kernel 2: depthwise conv (k=3/5/7/9) + groupnorm(4) + sigmoid -----
__global__ void k_sa(const float* __restrict__ xh, const float* __restrict__ xw,
                     const float* __restrict__ w3, const float* __restrict__ b3,
                     const float* __restrict__ w5, const float* __restrict__ b5,
                     const float* __restrict__ w7, const float* __restrict__ b7,
                     const float* __restrict__ w9, const float* __restrict__ b9,
                     const float* __restrict__ nhs, const float* __restrict__ nhb,
                     const float* __restrict__ nws, const float* __restrict__ nwb,
                     float* __restrict__ ha, float* __restrict__ wa) {
  int blk = blockIdx.x;            // b*8 + branch*4 + g
  int b = blk >> 3;
  int rem = blk & 7;
  int branch = rem >> 2;
  int g = rem & 3;
  int tid = threadIdx.x;

  const float* xm = branch ? xw : xh;
  const float* sc = branch ? nws : nhs;
  const float* bi = branch ? nwb : nhb;
  float* outp = branch ? wa : ha;

  const float* wk; const float* bk; int k;
  switch (g) {
    case 0: wk = w3; bk = b3; k = 3; break;
    case 1: wk = w5; bk = b5; k = 5; break;
    case 2: wk = w7; bk = b7; k = 7; break;
    default: wk = w9; bk = b9; k = 9; break;
  }
  int pad = k >> 1;
  const int NEL = 64 * LL;         // 3584 elements per group
  size_t base = ((size_t)b * C_ + g * 64) * LL;

  __shared__ float buf[64 * LL];
  __shared__ float rsum[128], rsq[128];
  __shared__ float smean, srstd;

  float s = 0.f, sq = 0.f;
  for (int e = tid; e < NEL; e += 128) {
    int cc = e / LL, l = e - cc * LL;
    float acc = bk[cc];
    const float* wr = wk + cc * k;
    const float* xr = xm + base + (size_t)cc * LL;
    for (int t = 0; t < k; ++t) {
      int li = l + t - pad;
      if (li >= 0 && li < LL) acc += wr[t] * xr[li];
    }
    buf[e] = acc;
    s += acc; sq += acc * acc;
  }
  rsum[tid] = s; rsq[tid] = sq;
  __syncthreads();
  for (int st = 64; st > 0; st >>= 1) {
    if (tid < st) { rsum[tid] += rsum[tid + st]; rsq[tid] += rsq[tid + st]; }
    __syncthreads();
  }
  if (tid == 0) {
    float mean = rsum[0] * (1.f / NEL);
    float var = rsq[0] * (1.f / NEL) - mean * mean;
    smean = mean; srstd = rsqrtf(var + EPSN);
  }
  __syncthreads();
  float mean = smean, rstd = srstd;
  for (int e = tid; e < NEL; e += 128) {
    int cc = e / LL, l = e - cc * LL;
    int c = g * 64 + cc;
    float vn = (buf[e] - mean) * rstd * sc[c] + bi[c];
    outp[((size_t)b * C_ + c) * LL + l] = sig_(vn);
  }
}

// ---- kernel 3: xg = x*ha*wa, 7x7 avg pool -> y_raw (B,C,8,8) -----------
// float4 streaming; 4 columns span at most 2 pooled cells.
__global__ void k_pool(const float* __restrict__ x,
                       const float* __restrict__ ha, const float* __restrict__ wa,
                       float* __restrict__ yraw) {
  int bc = blockIdx.x;
  const float4* p4 = reinterpret_cast<const float4*>(x + (size_t)bc * PL);
  const float* hr = ha + (size_t)bc * LL;
  const float4* wr4 = reinterpret_cast<const float4*>(wa + (size_t)bc * LL);
  __shared__ float pool[PN];
  int t = threadIdx.x;
  if (t < PN) pool[t] = 0.f;
  __syncthreads();
  for (int idx4 = t; idx4 < PL4; idx4 += 256) {
    float4 v = p4[idx4];
    int i = idx4 / ROW4;
    int jq = idx4 - i * ROW4;
    int j = jq * 4;
    float hi = hr[i];
    float4 w = wr4[jq];
    float a0 = v.x * hi * w.x;
    float a1 = v.y * hi * w.y;
    float a2 = v.z * hi * w.z;
    float a3 = v.w * hi * w.w;
    int pr = (i / 7) * PH;
    int cA = j / 7;
    int cB = (j + 3) / 7;
    float sA = a0, sB = 0.f;
    if ((j + 1) / 7 == cA) sA += a1; else sB += a1;
    if ((j + 2) / 7 == cA) sA += a2; else sB += a2;
    if (cB == cA) sA += a3; else sB += a3;
    atomicAdd(&pool[pr + cA], sA);
    if (cB != cA) atomicAdd(&pool[pr + cB], sB);
  }
  __syncthreads();
  if (t < PN) yraw[(size_t)bc * PN + t] = pool[t] * (1.f / 49.f);
}

// ---- kernel 4: per-batch groupnorm(1) over C*64 (float4) ---------------
__global__ void k_gn(const float* __restrict__ yraw,
                     const float* __restrict__ ns, const float* __restrict__ nb,
                     float* __restrict__ ynorm) {
  int b = blockIdx.x;
  int tid = threadIdx.x;
  const int N = C_ * PN;           // 16384
  const int N4 = N / 4;            // 4096
  const float4* p4 = reinterpret_cast<const float4*>(yraw + (size_t)b * N);
  float4* o4 = reinterpret_cast<float4*>(ynorm + (size_t)b * N);
  __shared__ float rsum[256], rsq[256];
  __shared__ float smean, srstd;
  float s = 0.f, sq = 0.f;
  for (int e = tid; e < N4; e += 256) {
    float4 v = p4[e];
    s += (v.x + v.y) + (v.z + v.w);
    sq += (v.x * v.x + v.y * v.y) + (v.z * v.z + v.w * v.w);
  }
  rsum[tid] = s; rsq[tid] = sq;
  __syncthreads();
  for (int st = 128; st > 0; st >>= 1) {
    if (tid < st) { rsum[tid] += rsum[tid + st]; rsq[tid] += rsq[tid + st]; }
    __syncthreads();
  }
  if (tid == 0) {
    float mean = rsum[0] * (1.f / N);
    float var = rsq[0] * (1.f / N) - mean * mean;
    smean = mean; srstd = rsqrtf(var + EPSN);
  }
  __syncthreads();
  float mean = smean, rstd = srstd;
  for (int e = tid; e < N4; e += 256) {
    int c = e >> 4;                 // 16 float4 per channel (PN/4)
    float scl = ns[c] * rstd, bia = nb[c] - mean * scl;
    float4 v = p4[e];
    float4 o;
    o.x = v.x * scl + bia;
    o.y = v.y * scl + bia;
    o.z = v.z * scl + bia;
    o.w = v.w * scl + bia;
    o4[e] = o;
  }
}

// ---- kernel 5: per-(b,h) attention with f32 WMMA; ca = sigmoid(mean) ---
__global__ void __launch_bounds__(32)
k_attn(const float* __restrict__ yn,
       const float* __restrict__ qw, const float* __restrict__ kw,
       const float* __restrict__ vw, float* __restrict__ ca) {
  int bh = blockIdx.x;
  int b = bh >> 3, h = bh & 7;
  int lane = threadIdx.x;
  __shared__ float qs[HD * PN];
  __shared__ float ks2[HD * PN];
  __shared__ float vs2[HD * PN];
  __shared__ float Sm[HD * HD];
  __shared__ float Om[HD * PN];

  const float4* y4 = reinterpret_cast<const float4*>(
      yn + ((size_t)b * C_ + h * HD) * PN);
  for (int idx4 = lane; idx4 < HD * PN / 4; idx4 += 32) {
    int d = idx4 >> 4;              // 16 float4 per row of 64
    int c = h * HD + d;
    float4 v = y4[idx4];
    float qc = qw[c], kc = kw[c], vc = vw[c];
    float4 q, k, vv;
    q.x = v.x * qc; q.y = v.y * qc; q.z = v.z * qc; q.w = v.w * qc;
    k.x = v.x * kc; k.y = v.y * kc; k.z = v.z * kc; k.w = v.w * kc;
    vv.x = v.x * vc; vv.y = v.y * vc; vv.z = v.z * vc; vv.w = v.w * vc;
    *reinterpret_cast<float4*>(&qs[idx4 * 4]) = q;
    *reinterpret_cast<float4*>(&ks2[idx4 * 4]) = k;
    *reinterpret_cast<float4*>(&vs2[idx4 * 4]) = vv;
  }
  __syncthreads();

  const float scale = 0.17677669529663687f;   // 1/sqrt(32)
  int hv = lane >> 4;                          // half-wave: selects K pair
  int lr = lane & 15;

  // S = Q * K^T   (four 16x16 tiles, contract over n=64 in steps of 4)
  for (int ti = 0; ti < 2; ++ti)
    for (int tj = 0; tj < 2; ++tj) {
      v8f acc = {};
      for (int n0 = 0; n0 < PN; n0 += 4) {
        int kk = n0 + 2 * hv;
        v2f a, bb;
        a.x = qs[(ti * 16 + lr) * PN + kk];
        a.y = qs[(ti * 16 + lr) * PN + kk + 1];
        bb.x = ks2[(tj * 16 + lr) * PN + kk];   // B = K^T: B[K=n][N=e]
        bb.y = ks2[(tj * 16 + lr) * PN + kk + 1];
        acc = wmma4(a, bb, acc);
      }
      for (int r = 0; r < 8; ++r)
        Sm[(ti * 16 + r + 8 * hv) * HD + tj * 16 + lr] = acc[r] * scale;
    }
  __syncthreads();

  // softmax over e (row-wise), one row per lane
  {
    int d = lane;
    float m = -1e30f;
    for (int e = 0; e < HD; ++e) m = fmaxf(m, Sm[d * HD + e]);
    float tmp[HD];
    float s = 0.f;
    for (int e = 0; e < HD; ++e) {
      float t = expf(Sm[d * HD + e] - m);
      tmp[e] = t; s += t;
    }
    float inv = 1.f / s;
    for (int e = 0; e < HD; ++e) Sm[d * HD + e] = tmp[e] * inv;
  }
  __syncthreads();

  // O = P * V   (eight 16x16 tiles, contract over e=32 in steps of 4)
  for (int ti = 0; ti < 2; ++ti)
    for (int tn = 0; tn < 4; ++tn) {
      v8f acc = {};
      for (int e0 = 0; e0 < HD; e0 += 4) {
        int kk = e0 + 2 * hv;
        v2f a, bb;
        a.x = Sm[(ti * 16 + lr) * HD + kk];
        a.y = Sm[(ti * 16 + lr) * HD + kk + 1];
        bb.x = vs2[kk * PN + tn * 16 + lr];     // B[K=e][N=n]
        bb.y = vs2[(kk + 1) * PN + tn * 16 + lr];
        acc = wmma4(a, bb, acc);
      }
      for (int r = 0; r < 8; ++r)
        Om[(ti * 16 + r + 8 * hv) * PN + tn * 16 + lr] = acc[r];
    }
  __syncthreads();

  // ca[c] = sigmoid(mean_n O[d][n]),  c = h*32 + d
  {
    int d = lane;
    float s = 0.f;
    for (int n = 0; n < PN; ++n) s += Om[d * PN + n];
    ca[(size_t)b * C_ + h * HD + d] = sig_(s * (1.f / PN));
  }
}

// ---- kernel 6: out = ca * x * ha * wa (pure float4 streaming) ----------
__global__ void k_final(const float* __restrict__ x,
                        const float* __restrict__ ha, const float* __restrict__ wa,
                        const float* __restrict__ ca, float* __restrict__ out) {
  int idx4 = blockIdx.x * 256 + threadIdx.x;
  const int TOT4 = B_ * C_ * PL4;             // 6,422,528
  if (idx4 >= TOT4) return;
  int bc = idx4 / PL4;
  int r = idx4 - bc * PL4;
  int i = r / ROW4;
  int jq = r - i * ROW4;
  float g = ca[bc] * ha[(size_t)bc * LL + i];
  const float4* x4 = reinterpret_cast<const float4*>(x);
  const float4* wa4 = reinterpret_cast<const float4*>(wa + (size_t)bc * LL);
  float4* o4 = reinterpret_cast<float4*>(out);
  float4 v = x4[idx4];
  float4 w = wa4[jq];
  float4 o;
  o.x = v.x * g * w.x;
  o.y = v.y * g * w.y;
  o.z = v.z * g * w.z;
  o.w = v.w * g * w.w;
  o4[idx4] = o;
}

// ---- launch ------------------------------------------------------------
extern "C" void kernel_launch(void* const* d_in, const int* in_sizes, int n_in,
                              void* d_out, int out_size, void* d_ws, size_t ws_size,
                              hipStream_t stream) {
  const float* x   = (const float*)d_in[0];
  const float* w3  = (const float*)d_in[1];
  const float* b3  = (const float*)d_in[2];
  const float* w5  = (const float*)d_in[3];
  const float* b5  = (const float*)d_in[4];
  const float* w7  = (const float*)d_in[5];
  const float* b7  = (const float*)d_in[6];
  const float* w9  = (const float*)d_in[7];
  const float* b9  = (const float*)d_in[8];
  const float* nhs = (const float*)d_in[9];
  const float* nhb = (const float*)d_in[10];
  const float* nws = (const float*)d_in[11];
  const float* nwb = (const float*)d_in[12];
  const float* ns  = (const float*)d_in[13];
  const float* nb  = (const float*)d_in[14];
  const float* qw  = (const float*)d_in[15];
  const float* kw  = (const float*)d_in[16];
  const float* vw  = (const float*)d_in[17];
  float* out = (float*)d_out;

  float* wsf = (float*)d_ws;
  const size_t NBC = (size_t)B_ * C_;        // 8192
  float* xh    = wsf;                        // B*C*56
  float* xw    = xh + NBC * LL;
  float* ha    = xw + NBC * LL;
  float* wa    = ha + NBC * LL;
  float* yraw  = wa + NBC * LL;              // B*C*64
  float* ynorm = yraw + NBC * PN;
  float* ca    = ynorm + NBC * PN;           // B*C

  k_means<<<B_ * C_, 256, 0, stream>>>(x, xh, xw);
  k_sa<<<B_ * 2 * 4, 128, 0, stream>>>(xh, xw, w3, b3, w5, b5, w7, b7, w9, b9,
                                       nhs, nhb, nws, nwb, ha, wa);
  k_pool<<<B_ * C_, 256, 0, stream>>>(x, ha, wa, yraw);
  k_gn<<<B_, 256, 0, stream>>>(yraw, ns, nb, ynorm);
  k_attn<<<B_ * HEADS_, 32, 0, stream>>>(ynorm, qw, kw, vw, ca);
  const int TOT4 = B_ * C_ * PL4;
  k_final<<<(TOT4 + 255) / 256, 256, 0, stream>>>(x, ha, wa, ca, out);
}